// GPSAttentionLayer_88776974009099
// MI455X (gfx1250) — compile-verified
//
#include <hip/hip_runtime.h>
#include <hip/hip_bf16.h>
#include <float.h>

#define NNODES 20000
#define FDIM 256
#define RNEI 32
#define BN_EPS 1e-5f
#define MASKV (-1e30f)

typedef float v2f __attribute__((ext_vector_type(2)));
typedef float v8f __attribute__((ext_vector_type(8)));

// ---------------------------------------------------------------------------
// K1: Key = input @ Wk, Query = input @ Wq   (ATT_HID == 1 -> row dot products)
// One wave (32 lanes) per node row.
// ---------------------------------------------------------------------------
__global__ void keyquery_kernel(const float* __restrict__ in,
                                const float* __restrict__ Wk,
                                const float* __restrict__ Wq,
                                float* __restrict__ key,
                                float* __restrict__ query, int n) {
  int wave = (blockIdx.x * blockDim.x + threadIdx.x) >> 5;
  int lane = threadIdx.x & 31;
  if (wave >= n) return;
  const float* row = in + (size_t)wave * FDIM;
  float k = 0.f, q = 0.f;
  #pragma unroll
  for (int c = 0; c < FDIM; c += 32) {
    float x = row[c + lane];
    k += x * Wk[c + lane];
    q += x * Wq[c + lane];
  }
  #pragma unroll
  for (int off = 16; off > 0; off >>= 1) {
    k += __shfl_down(k, off, 32);
    q += __shfl_down(q, off, 32);
  }
  if (lane == 0) { key[wave] = k; query[wave] = q; }
}

// ---------------------------------------------------------------------------
// K2: new_h = input[N,256] @ W[256,256] using V_WMMA_F32_16X16X4_F32.
// One wave owns a 16x64 output tile (4 accumulators, A fragment reused).
// f32 16x4 A layout : lane(0..15|16..31) holds row M=lane&15, K = khalf+{0,1}
// f32 4x16 B layout : VGPR v holds B[khalf+v][lane&15]
// f32 16x16 C layout: VGPR v holds D[v + 8*(lane>>4)][lane&15]
// ---------------------------------------------------------------------------
__global__ void gemm_wmma_kernel(const float* __restrict__ A,
                                 const float* __restrict__ W,
                                 float* __restrict__ D, int n) {
  const int wv    = threadIdx.x >> 5;
  const int lane  = threadIdx.x & 31;
  const int mbase = blockIdx.x * 64 + wv * 16;       // wave-uniform
  if (mbase + 16 > n) return;                        // uniform branch: EXEC stays all-1
  const int nbase = blockIdx.y * 64;
  const int mrow  = mbase + (lane & 15);
  const int khalf = (lane >> 4) * 2;                 // 0 or 2
  const int ncol  = lane & 15;

  v8f c0 = {}, c1 = {}, c2 = {}, c3 = {};
  for (int k0 = 0; k0 < FDIM; k0 += 4) {
    const float* ap = A + (size_t)mrow * FDIM + k0 + khalf;
    v2f a; a.x = ap[0]; a.y = ap[1];
    const float* wp = W + (size_t)(k0 + khalf) * FDIM + nbase + ncol;
    v2f b0; b0.x = wp[0];  b0.y = wp[FDIM + 0];
    v2f b1; b1.x = wp[16]; b1.y = wp[FDIM + 16];
    v2f b2; b2.x = wp[32]; b2.y = wp[FDIM + 32];
    v2f b3; b3.x = wp[48]; b3.y = wp[FDIM + 48];
    c0 = __builtin_amdgcn_wmma_f32_16x16x4_f32(false, a, false, b0, (short)0, c0, false, false);
    c1 = __builtin_amdgcn_wmma_f32_16x16x4_f32(false, a, false, b1, (short)0, c1, false, false);
    c2 = __builtin_amdgcn_wmma_f32_16x16x4_f32(false, a, false, b2, (short)0, c2, false, false);
    c3 = __builtin_amdgcn_wmma_f32_16x16x4_f32(false, a, false, b3, (short)0, c3, false, false);
  }
  const int rbase = mbase + (lane >> 4) * 8;
  #pragma unroll
  for (int v = 0; v < 8; ++v) {
    float* dp = D + (size_t)(rbase + v) * FDIM + nbase + ncol;
    dp[0]  = c0[v];
    dp[16] = c1[v];
    dp[32] = c2[v];
    dp[48] = c3[v];
  }
}

// ---------------------------------------------------------------------------
// K3: per-node masked softmax over the 32 neighbors + weighted row gather.
// top_k(k=R)+softmax+gather-sum is permutation invariant -> no sort needed.
// Masked entries (-1e30) underflow exp() to exactly 0, matching min-1000.
// ---------------------------------------------------------------------------
__global__ void attn_aggregate_kernel(const float* __restrict__ newh,
                                      const float* __restrict__ key,
                                      const float* __restrict__ query,
                                      const int* __restrict__ rf,
                                      float* __restrict__ prebn, int n) {
  __shared__ float w_s[RNEI];
  __shared__ int   nb_s[RNEI];
  const int i   = blockIdx.x;
  const int tid = threadIdx.x;
  if (tid < RNEI) {                       // first wave, lanes 0..31
    int nb = rf[i * RNEI + tid];
    float att = key[i] * query[nb];
    if (nb == n - 1) att = MASKV;
    float m = att;
    #pragma unroll
    for (int off = 16; off > 0; off >>= 1) m = fmaxf(m, __shfl_xor(m, off, 32));
    float e = expf(att - m);
    float s = e;
    #pragma unroll
    for (int off = 16; off > 0; off >>= 1) s += __shfl_xor(s, off, 32);
    w_s[tid]  = e / s;
    nb_s[tid] = nb;
  }
  __syncthreads();
  float acc = newh[(size_t)i * FDIM + tid];          // self term: final_h = new_h + ...
  #pragma unroll 4
  for (int j = 0; j < RNEI; ++j) {
    if (j + 1 < RNEI)
      __builtin_prefetch(&newh[(size_t)nb_s[j + 1] * FDIM + tid], 0, 1);
    acc += w_s[j] * newh[(size_t)nb_s[j] * FDIM + tid];
  }
  prebn[(size_t)i * FDIM + tid] = acc;
}

// ---------------------------------------------------------------------------
// K4: BatchNorm batch statistics (population mean/var per feature).
// One block per feature column.
// ---------------------------------------------------------------------------
__global__ void bn_stats_kernel(const float* __restrict__ prebn,
                                float* __restrict__ meanv,
                                float* __restrict__ varv, int n) {
  __shared__ float ssum[256];
  __shared__ float ssq[256];
  const int f = blockIdx.x;
  float s = 0.f, q = 0.f;
  for (int r = threadIdx.x; r < n; r += blockDim.x) {
    float x = prebn[(size_t)r * FDIM + f];
    s += x; q += x * x;
  }
  ssum[threadIdx.x] = s; ssq[threadIdx.x] = q;
  __syncthreads();
  for (int off = 128; off > 0; off >>= 1) {
    if (threadIdx.x < off) {
      ssum[threadIdx.x] += ssum[threadIdx.x + off];
      ssq[threadIdx.x]  += ssq[threadIdx.x + off];
    }
    __syncthreads();
  }
  if (threadIdx.x == 0) {
    float mean = ssum[0] / (float)n;
    meanv[f] = mean;
    varv[f]  = ssq[0] / (float)n - mean * mean;
  }
}

// ---------------------------------------------------------------------------
// K5: apply BN (affine) + ReLU in place on d_out's float region.
// ---------------------------------------------------------------------------
__global__ void bn_apply_relu_kernel(float* __restrict__ io,
                                     const float* __restrict__ meanv,
                                     const float* __restrict__ varv,
                                     const float* __restrict__ gamma,
                                     const float* __restrict__ beta) {
  const int idx = blockIdx.x * blockDim.x + threadIdx.x;
  const int f = idx & (FDIM - 1);
  float x = io[idx];
  float y = gamma[f] * (x - meanv[f]) * rsqrtf(varv[f] + BN_EPS) + beta[f];
  io[idx] = fmaxf(y, 0.f);
}

// ---------------------------------------------------------------------------
// K6: 2-hop expansion. Per node: 1024 candidates adj[rf[i,j]][l], score with
// Key_i*Query_c, mask padding with -1e30 (strictly below all real scores;
// stable index tie-break reproduces jax.lax.top_k on the masked min-1 value),
// then 32 rounds of stable arg-max selection in LDS. Also copies rf -> out.
// ---------------------------------------------------------------------------
__global__ void expand_kernel(const float* __restrict__ key,
                              const float* __restrict__ query,
                              const int* __restrict__ rf,
                              const int* __restrict__ adj,
                              int* __restrict__ out_rf, int n) {
  __shared__ float att_s[1024];
  __shared__ int   cand_s[1024];
  __shared__ float sval[256];
  __shared__ int   sidx[256];
  const int i   = blockIdx.x;
  const int tid = threadIdx.x;
  const float ki = key[i];

  if (tid < RNEI) out_rf[(size_t)i * RNEI + tid] = rf[i * RNEI + tid];

  #pragma unroll
  for (int q = 0; q < 4; ++q) {
    int c = q * 256 + tid;
    int j = c >> 5, l = c & 31;
    int mid = rf[i * RNEI + j];
    int nb  = adj[mid * RNEI + l];
    float a = ki * query[nb];
    if (nb == n - 1) a = MASKV;
    cand_s[c] = nb;
    att_s[c]  = a;
  }
  __syncthreads();

  for (int s = 0; s < RNEI; ++s) {
    float bv = -FLT_MAX; int bi = 1 << 30;
    #pragma unroll
    for (int q = 0; q < 4; ++q) {
      int c = q * 256 + tid;
      float a = att_s[c];
      if (a > bv || (a == bv && c < bi)) { bv = a; bi = c; }
    }
    sval[tid] = bv; sidx[tid] = bi;
    __syncthreads();
    for (int off = 128; off > 0; off >>= 1) {
      if (tid < off) {
        float ov = sval[tid + off]; int oi = sidx[tid + off];
        if (ov > sval[tid] || (ov == sval[tid] && oi < sidx[tid])) {
          sval[tid] = ov; sidx[tid] = oi;
        }
      }
      __syncthreads();
    }
    if (tid == 0) {
      int b = sidx[0];
      out_rf[(size_t)n * RNEI + (size_t)i * RNEI + s] = cand_s[b];
      att_s[b] = -FLT_MAX;   // consume (below mask value -> never re-picked)
    }
    __syncthreads();
  }
}

// ---------------------------------------------------------------------------
extern "C" void kernel_launch(void* const* d_in, const int* in_sizes, int n_in,
                              void* d_out, int out_size, void* d_ws, size_t ws_size,
                              hipStream_t stream) {
  const float* input = (const float*)d_in[0];   // [N, 256]
  const int*   rf    = (const int*)d_in[1];     // [1, N, 32]
  const int*   adj   = (const int*)d_in[2];     // [N, 32]
  const float* W     = (const float*)d_in[3];   // [1, 256, 256]
  const float* Wk    = (const float*)d_in[4];   // [256, 1]
  const float* Wq    = (const float*)d_in[5];   // [256, 1]
  const float* gamma = (const float*)d_in[6];   // [256]
  const float* beta  = (const float*)d_in[7];   // [256]

  const int n = NNODES;

  // workspace layout (floats)
  float* key   = (float*)d_ws;
  float* query = key + n;
  float* newh  = query + n;                     // [N, 256]
  float* meanv = newh + (size_t)n * FDIM;       // [256]
  float* varv  = meanv + FDIM;                  // [256]

  float* prebn  = (float*)d_out;                              // final_h region
  int*   out_rf = (int*)((float*)d_out + (size_t)n * FDIM);   // new_receptive_field

  // K1: Key / Query
  keyquery_kernel<<<(n * 32 + 255) / 256, 256, 0, stream>>>(input, Wk, Wq, key, query, n);

  // K2: new_h = input @ W  (WMMA f32)
  dim3 ggrid((n + 63) / 64, FDIM / 64);
  gemm_wmma_kernel<<<ggrid, 128, 0, stream>>>(input, W, newh, n);

  // K3: softmax-weighted neighbor aggregation -> pre-BN final_h (in d_out)
  attn_aggregate_kernel<<<n, FDIM, 0, stream>>>(newh, key, query, rf, prebn, n);

  // K4: batch statistics
  bn_stats_kernel<<<FDIM, 256, 0, stream>>>(prebn, meanv, varv, n);

  // K5: BN + ReLU in place
  bn_apply_relu_kernel<<<((size_t)n * FDIM) / 256, 256, 0, stream>>>(prebn, meanv, varv, gamma, beta);

  // K6: 2-hop receptive-field expansion (+ copy of rf)
  expand_kernel<<<n, 256, 0, stream>>>(key, query, rf, adj, out_rf, n);
}